// RadarEncV7_35450660061779
// MI455X (gfx1250) — compile-verified
//
#include <hip/hip_runtime.h>
#include <hip/hip_bf16.h>
#include <climits>

// ---------------- types / constants ----------------
typedef float v2f __attribute__((ext_vector_type(2)));
typedef float v8f __attribute__((ext_vector_type(8)));

#define NPTS   43750
#define NRAYS  175
#define KPTS   250
#define TBITS  18
#define TSIZE  (1 << TBITS)
#define TMASK  (TSIZE - 1)
#define M0CAP  43750
#define MCAP   49152            // caps for levels 1..3 (idx/feature arrays)
#define EMPTYKEY 0xFFFFFFFFFFFFFFFFull
#define OUTN   (256 * 128 * 128 * 14)

static __device__ __forceinline__ int dmin(int a, int b) { return a < b ? a : b; }

static __device__ __forceinline__ unsigned hash_key(unsigned long long k) {
  return (unsigned)((k * 0x9E3779B97F4A7C15ull) >> (64 - TBITS)) & TMASK;
}

static __device__ void table_insert(unsigned long long* keys, int* vals,
                                    unsigned long long key, int row) {
  unsigned h = hash_key(key);
  for (int p = 0; p < TSIZE; ++p) {
    unsigned long long prev = atomicCAS(&keys[h], EMPTYKEY, key);
    if (prev == EMPTYKEY || prev == key) { atomicMin(&vals[h], row); return; }
    h = (h + 1) & TMASK;
  }
}

static __device__ int table_lookup(const unsigned long long* keys, const int* vals,
                                   unsigned long long key) {
  unsigned h = hash_key(key);
  for (int p = 0; p < TSIZE; ++p) {
    unsigned long long cur = keys[h];
    if (cur == key) return vals[h];
    if (cur == EMPTYKEY) return -1;
    h = (h + 1) & TMASK;
  }
  return -1;
}

// ---------------- init / zero ----------------
__global__ void init_tables_kernel(unsigned long long* keys, int* vals, int n,
                                   int* cnts, int ncnt) {
  int i = blockIdx.x * blockDim.x + threadIdx.x;
  if (i < n) { keys[i] = EMPTYKEY; vals[i] = INT_MAX; }
  if (i < ncnt) cnts[i] = 0;
}

__global__ void zero_f32_kernel(float* p, long long n) {
  long long i = (long long)blockIdx.x * blockDim.x + threadIdx.x;
  if (i < n) p[i] = 0.f;
}

// ---------------- attention (per-ray, VALU+LDS) ----------------
__global__ __launch_bounds__(256) void attn_kernel(
    const float* __restrict__ vf, const int* __restrict__ coors,
    const float* __restrict__ ele_emb, const float* __restrict__ azi_emb,
    const float* __restrict__ proj_w, const float* __restrict__ proj_b,
    const float* __restrict__ wq, const float* __restrict__ bq,
    const float* __restrict__ wk, const float* __restrict__ bk,
    const float* __restrict__ wv, const float* __restrict__ bv,
    float* __restrict__ aout) {
  int r = blockIdx.x;
  int t = threadIdx.x;
  __shared__ float sx[KPTS][8];
  __shared__ float see[KPTS][3];
  __shared__ float sae[KPTS][3];
  __shared__ float sk[KPTS][32];
  __shared__ float sv[KPTS][8];
  __shared__ float swq[14 * 32];
  __shared__ float swk[14 * 32];
  __shared__ float swv[14 * 8];
  __shared__ float sbq[32], sbk[32], sbv[8];

  if (t < KPTS) {
    int p = r * KPTS + t;
    float f8[8];
    for (int c = 0; c < 8; ++c) f8[c] = vf[(size_t)p * 8 + c];
    for (int c = 3; c < 6; ++c) f8[c] = (float)(int)f8[c];
    int ei = coors[(size_t)p * 4 + 1];
    int ai = coors[(size_t)p * 4 + 3];
    for (int c = 0; c < 3; ++c) { see[t][c] = ele_emb[ei * 3 + c]; sae[t][c] = azi_emb[ai * 3 + c]; }
    for (int o = 0; o < 8; ++o) {
      float acc = proj_b[o];
      for (int c = 0; c < 8; ++c) acc += f8[c] * proj_w[c * 8 + o];
      sx[t][o] = acc;
    }
  }
  __syncthreads();

  for (int l = 0; l < 4; ++l) {
    for (int i = t; i < 14 * 32; i += 256) { swq[i] = wq[l * 14 * 32 + i]; swk[i] = wk[l * 14 * 32 + i]; }
    for (int i = t; i < 14 * 8; i += 256) swv[i] = wv[l * 14 * 8 + i];
    if (t < 32) { sbq[t] = bq[l * 32 + t]; sbk[t] = bk[l * 32 + t]; }
    if (t < 8) sbv[t] = bv[l * 8 + t];
    __syncthreads();

    float qreg[32];
    if (t < KPTS) {
      float xi[14];
      for (int c = 0; c < 8; ++c) xi[c] = sx[t][c];
      for (int c = 0; c < 3; ++c) { xi[8 + c] = see[t][c]; xi[11 + c] = sae[t][c]; }
      for (int o = 0; o < 32; ++o) {
        float aq = sbq[o], ak = sbk[o];
        for (int c = 0; c < 14; ++c) { aq += xi[c] * swq[c * 32 + o]; ak += xi[c] * swk[c * 32 + o]; }
        qreg[o] = aq; sk[t][o] = ak;
      }
      for (int o = 0; o < 8; ++o) {
        float av = sbv[o];
        for (int c = 0; c < 14; ++c) av += xi[c] * swv[c * 8 + o];
        sv[t][o] = av;
      }
    }
    __syncthreads();

    float outv[8] = {0, 0, 0, 0, 0, 0, 0, 0};
    if (t < KPTS) {
      const float scale = 0.17677669529663687f;  // 1/sqrt(32)
      float mmax = -3.4e38f;
      for (int j = 0; j < KPTS; ++j) {
        float s = 0.f;
        for (int c = 0; c < 32; ++c) s += qreg[c] * sk[j][c];
        s *= scale;
        mmax = fmaxf(mmax, s);
      }
      float sum = 0.f;
      for (int j = 0; j < KPTS; ++j) {
        float s = 0.f;
        for (int c = 0; c < 32; ++c) s += qreg[c] * sk[j][c];
        float e = expf(s * scale - mmax);
        sum += e;
        for (int c = 0; c < 8; ++c) outv[c] += e * sv[j][c];
      }
      float inv = 1.f / sum;
      for (int c = 0; c < 8; ++c) outv[c] *= inv;
    }
    __syncthreads();
    if (t < KPTS)
      for (int c = 0; c < 8; ++c) sx[t][c] = outv[c];
    __syncthreads();
  }

  if (t < KPTS)
    for (int c = 0; c < 8; ++c) aout[((size_t)r * KPTS + t) * 8 + c] = sx[t][c];
}

// ---------------- voxelize + dedup ----------------
__global__ void voxelize_kernel(const int* __restrict__ coors,
                                unsigned long long* keys0, int* vals0) {
  int i = blockIdx.x * blockDim.x + threadIdx.x;
  if (i >= NPTS) return;
  const float d2r = 0.017453292519943295f;
  float rr = (float)coors[(size_t)i * 4 + 2] * 0.46f;
  float el = ((float)coors[(size_t)i * 4 + 1] - 18.0f) * d2r;
  float az = ((float)coors[(size_t)i * 4 + 3] - 53.0f) * d2r;
  float ce = cosf(el);
  float x = rr * ce * cosf(az);
  float y = rr * ce * sinf(az);
  float z = rr * sinf(el);
  if (!(x >= 0.f && x <= 51.2f && y >= -25.6f && y <= 25.6f && z >= -2.6f && z <= 3.0f)) return;
  long long gz = (long long)floorf((z + 2.6f) / 0.05f);
  long long gy = (long long)floorf((y + 25.6f) / 0.05f);
  long long gx = (long long)floorf(x / 0.05f);
  unsigned long long key = (unsigned long long)((gz * 1024 + gy) * 1024 + gx);
  table_insert(keys0, vals0, key, i);
}

// ---------------- compaction ----------------
__global__ void compact_table_kernel(unsigned long long* keys, int* vals,
                                     int SY, int SX, int* uc, int* rows_out,
                                     int* cnt, int mcap) {
  int s = blockIdx.x * blockDim.x + threadIdx.x;
  if (s >= TSIZE) return;
  unsigned long long k = keys[s];
  if (k == EMPTYKEY) return;
  int pos = atomicAdd(cnt, 1);
  if (pos < mcap) {
    long long kk = (long long)k;
    int x = (int)(kk % SX);
    long long r = kk / SX;
    int y = (int)(r % SY);
    int z = (int)(r / SY);
    uc[pos * 3 + 0] = z; uc[pos * 3 + 1] = y; uc[pos * 3 + 2] = x;
    if (rows_out) rows_out[pos] = vals[s];
    vals[s] = pos;
  } else {
    vals[s] = -1;
  }
}

// ---------------- strided output-voxel generation ----------------
__global__ void strided_insert_kernel(const int* __restrict__ uc, const int* Mptr, int mcap,
                                      unsigned long long* keys, int* vals,
                                      int OZ, int OY, int OX) {
  int m = blockIdx.x * blockDim.x + threadIdx.x;
  int M = dmin(*Mptr, mcap);
  if (m >= M) return;
  int cz = uc[m * 3], cy = uc[m * 3 + 1], cx = uc[m * 3 + 2];
  for (int kk = 0; kk < 27; ++kk) {
    int dz = kk / 9 - 1, dy = (kk / 3) % 3 - 1, dx = kk % 3 - 1;
    int tz = cz - dz, ty = cy - dy, tx = cx - dx;
    if (tz < 0 || ty < 0 || tx < 0) continue;
    if ((tz & 1) || (ty & 1) || (tx & 1)) continue;
    int oz = tz >> 1, oy = ty >> 1, ox = tx >> 1;
    if (oz >= OZ || oy >= OY || ox >= OX) continue;
    unsigned long long key =
        ((unsigned long long)oz * OY + (unsigned long long)oy) * (unsigned long long)OX + ox;
    table_insert(keys, vals, key, 0);
  }
}

// ---------------- rulebook (27-neighbor index tables) ----------------
__global__ void build_idx27_kernel(const int* __restrict__ uc, const int* Mptr, int mcap,
                                   const unsigned long long* __restrict__ keysQ,
                                   const int* __restrict__ valsQ,
                                   int QZ, int QY, int QX, int scale,
                                   int* __restrict__ idx_out) {
  int m = blockIdx.x * blockDim.x + threadIdx.x;
  int M = dmin(*Mptr, mcap);
  if (m >= M) return;
  int cz = uc[m * 3] * scale, cy = uc[m * 3 + 1] * scale, cx = uc[m * 3 + 2] * scale;
  for (int kk = 0; kk < 27; ++kk) {
    int dz = kk / 9 - 1, dy = (kk / 3) % 3 - 1, dx = kk % 3 - 1;
    int nz = cz + dz, ny = cy + dy, nx = cx + dx;
    int idx = -1;
    if (nz >= 0 && ny >= 0 && nx >= 0 && nz < QZ && ny < QY && nx < QX) {
      unsigned long long key =
          ((unsigned long long)nz * QY + (unsigned long long)ny) * (unsigned long long)QX + nx;
      idx = table_lookup(keysQ, valsQ, key);
    }
    idx_out[(size_t)kk * MCAP + m] = idx;
  }
}

// ---------------- input layer: [geo|attn] @ in_w + GN(8) + relu ----------------
__global__ void in_layer_kernel(const int* __restrict__ uc0, const int* __restrict__ rows0,
                                const float* __restrict__ a,
                                const float* __restrict__ in_w, const float* __restrict__ in_b,
                                const float* __restrict__ in_g, const float* __restrict__ in_be,
                                const int* Mptr, int mcap, float* __restrict__ fout) {
  int m = blockIdx.x * blockDim.x + threadIdx.x;
  int M = dmin(*Mptr, mcap);
  if (m >= M) return;
  float in[11];
  in[0] = (float)uc0[m * 3 + 0];
  in[1] = (float)uc0[m * 3 + 1];
  in[2] = (float)uc0[m * 3 + 2];
  const float* ar = a + (size_t)rows0[m] * 8;
  for (int c = 0; c < 8; ++c) in[3 + c] = ar[c];
  float y[16];
  for (int o = 0; o < 16; ++o) {
    float acc = in_b[o];
    for (int c = 0; c < 11; ++c) acc += in[c] * in_w[c * 16 + o];
    y[o] = acc;
  }
  for (int g = 0; g < 8; ++g) {
    float a0 = y[2 * g], a1 = y[2 * g + 1];
    float mu = 0.5f * (a0 + a1);
    float v = 0.5f * ((a0 - mu) * (a0 - mu) + (a1 - mu) * (a1 - mu));
    float rs = rsqrtf(v + 1e-5f);
    float o0 = (a0 - mu) * rs * in_g[2 * g] + in_be[2 * g];
    float o1 = (a1 - mu) * rs * in_g[2 * g + 1] + in_be[2 * g + 1];
    fout[(size_t)m * 16 + 2 * g] = fmaxf(o0, 0.f);
    fout[(size_t)m * 16 + 2 * g + 1] = fmaxf(o1, 0.f);
  }
}

// ---------------- WMMA sparse conv ----------------
// One 128-thread workgroup (4 waves) per 16-row output tile.
// Per tap: 16xCIN activation tile gathered ONCE into LDS via async global->LDS
// (ASYNCcnt, s_wait_asynccnt), each wave computes one 16-column tile from it
// with v_wmma_f32_16x16x4_f32 (A read from LDS as 8B ds_loads, B from L2-resident
// weights with global_prefetch of the next tap).
template <int CIN, int COUT>
__global__ __launch_bounds__(128) void conv_wmma_kernel(
    const float* __restrict__ fin, const float* __restrict__ W,
    const int* __restrict__ idx27, int idx_stride,
    const int* Mptr, int mcap, float* __restrict__ fout) {
  __shared__ float sA[16 * CIN];
  __shared__ int sIdx[16];
  int M = dmin(*Mptr, mcap);
  int tile = blockIdx.x * 16;
  if (tile >= M) return;  // uniform: no barrier before this point
  int tid = threadIdx.x;
  int wave = tid >> 5;
  int lane = tid & 31;
  int ctile = blockIdx.y * 4 + wave;
  bool active = (ctile * 16) < COUT;  // wave-uniform
  int cout0 = ctile * 16;
  int kbase = (lane >> 4) << 1;  // A/B 16x4 layout: lanes 0-15 K={0,1}, 16-31 K={2,3}
  int colB = lane & 15;
  int rowA = lane & 15;
  v8f acc = {0.f, 0.f, 0.f, 0.f, 0.f, 0.f, 0.f, 0.f};

  for (int k27 = 0; k27 < 27; ++k27) {
    if (tid < 16) {
      int gm = tile + tid;
      sIdx[tid] = (gm < M) ? idx27[(size_t)k27 * idx_stride + gm] : -1;
    }
    __syncthreads();
    // cooperative gather: 16 rows x CIN floats, 16B granules
    constexpr int N4 = (16 * CIN) / 4;
#pragma unroll
    for (int e0 = 0; e0 < N4; e0 += 128) {
      int e = e0 + tid;
      if (e < N4) {
        int row = (e * 4) / CIN;
        int c = (e * 4) % CIN;
        int src = sIdx[row];
        float* dst = &sA[row * CIN + c];
        if (src >= 0) {
          unsigned ldsoff = (unsigned)(size_t)dst;
          unsigned long long gaddr = (unsigned long long)(size_t)(fin + (size_t)src * CIN + c);
          asm volatile("global_load_async_to_lds_b128 %0, %1, off"
                       :: "v"(ldsoff), "v"(gaddr) : "memory");
        } else {
          dst[0] = 0.f; dst[1] = 0.f; dst[2] = 0.f; dst[3] = 0.f;
        }
      }
    }
    asm volatile("s_wait_asynccnt 0x0" ::: "memory");
    __syncthreads();
    if (active) {
      const float* wk = W + (size_t)k27 * CIN * COUT + cout0;
      if (k27 + 1 < 27)
        __builtin_prefetch(W + (size_t)(k27 + 1) * CIN * COUT + cout0, 0, 3);
#pragma unroll
      for (int c0 = 0; c0 < CIN; c0 += 4) {
        v2f a = *(const v2f*)(&sA[rowA * CIN + c0 + kbase]);
        v2f b;
        b.x = wk[(size_t)(c0 + kbase) * COUT + colB];
        b.y = wk[(size_t)(c0 + kbase + 1) * COUT + colB];
        acc = __builtin_amdgcn_wmma_f32_16x16x4_f32(false, a, false, b, (short)0, acc,
                                                    false, false);
      }
    }
    __syncthreads();
  }
  if (active) {
    // C/D layout: VGPR v, lanes 0-15 -> (M=v, N=lane); lanes 16-31 -> (M=v+8, N=lane-16)
    int half = lane >> 4;
    int col = lane & 15;
#pragma unroll
    for (int v = 0; v < 8; ++v) {
      int gr = tile + v + half * 8;
      if (gr < M) fout[(size_t)gr * COUT + cout0 + col] = acc[v];
    }
  }
}

// ---------------- batchnorm: stats then apply (+res, +relu) ----------------
__global__ __launch_bounds__(256) void bn_stats_kernel(const float* __restrict__ f,
                                                       const int* Mptr, int mcap, int C,
                                                       float* mean, float* var) {
  int c = blockIdx.x;
  int M = dmin(*Mptr, mcap);
  float s = 0.f, s2 = 0.f;
  for (int m = threadIdx.x; m < M; m += 256) {
    float v = f[(size_t)m * C + c];
    s += v; s2 += v * v;
  }
  __shared__ float rs[256], rs2[256];
  rs[threadIdx.x] = s; rs2[threadIdx.x] = s2;
  __syncthreads();
  for (int o = 128; o > 0; o >>= 1) {
    if (threadIdx.x < o) { rs[threadIdx.x] += rs[threadIdx.x + o]; rs2[threadIdx.x] += rs2[threadIdx.x + o]; }
    __syncthreads();
  }
  if (threadIdx.x == 0) {
    float inv = (M > 0) ? 1.f / (float)M : 0.f;
    float mu = rs[0] * inv;
    mean[c] = mu;
    var[c] = rs2[0] * inv - mu * mu;
  }
}

__global__ void bn_apply_kernel(const float* __restrict__ x, const float* __restrict__ g,
                                const float* __restrict__ be, const float* __restrict__ mean,
                                const float* __restrict__ var, const float* __restrict__ res,
                                float* __restrict__ out, const int* Mptr, int mcap, int C,
                                int relu) {
  long long i = (long long)blockIdx.x * blockDim.x + threadIdx.x;
  int M = dmin(*Mptr, mcap);
  if (i >= (long long)M * C) return;
  int c = (int)(i % C);
  float v = (x[i] - mean[c]) * rsqrtf(var[c] + 1e-5f) * g[c] + be[c];
  if (res) v += res[i];
  if (relu) v = fmaxf(v, 0.f);
  out[i] = v;
}

// ---------------- output GN(16 groups over 256) + relu (in place) ----------------
__global__ __launch_bounds__(256) void gn_out_kernel(float* __restrict__ f,
                                                     const float* __restrict__ ob,
                                                     const float* __restrict__ og,
                                                     const float* __restrict__ obe,
                                                     const int* Mptr, int mcap) {
  int m = blockIdx.x;
  int c = threadIdx.x;
  int M = dmin(*Mptr, mcap);
  if (m >= M) return;
  __shared__ float sv[256];
  float v = f[(size_t)m * 256 + c] + ob[c];
  sv[c] = v;
  __syncthreads();
  int g0 = (c >> 4) << 4;
  float mu = 0.f;
  for (int i = 0; i < 16; ++i) mu += sv[g0 + i];
  mu *= (1.f / 16.f);
  float var = 0.f;
  for (int i = 0; i < 16; ++i) { float d = sv[g0 + i] - mu; var += d * d; }
  var *= (1.f / 16.f);
  float y = (v - mu) * rsqrtf(var + 1e-5f) * og[c] + obe[c];
  f[(size_t)m * 256 + c] = fmaxf(y, 0.f);
}

// ---------------- dense scatter: out[1,C,X,Y,Z] ----------------
__global__ __launch_bounds__(256) void scatter_kernel(const float* __restrict__ f,
                                                      const int* __restrict__ uc3,
                                                      const int* Mptr, int mcap,
                                                      float* __restrict__ out) {
  int m = blockIdx.x;
  int c = threadIdx.x;
  int M = dmin(*Mptr, mcap);
  if (m >= M) return;
  int z = uc3[m * 3 + 0], y = uc3[m * 3 + 1], x = uc3[m * 3 + 2];
  out[(((size_t)c * 128 + x) * 128 + y) * 14 + z] = f[(size_t)m * 256 + c];
}

// ---------------- host-side stage driver ----------------
template <int CIN, int COUT>
static void run_stage(const float* cw, const float* cg, const float* cbe,
                      const float* bw1, const float* bg1, const float* bbe1,
                      const float* bw2, const float* bg2, const float* bbe2,
                      float** curp, float** altp, float* FH,
                      const int* strI, const int* subI, const int* cnt,
                      float* MEAN, float* VAR, hipStream_t stream) {
  const int TPB = 256;
  float* cur = *curp;
  float* alt = *altp;
  dim3 cgrid(MCAP / 16, (COUT + 63) / 64);
  // strided conv: cur(level s-1) -> alt(level s)
  conv_wmma_kernel<CIN, COUT><<<cgrid, 128, 0, stream>>>(cur, cw, strI, MCAP, cnt, MCAP, alt);
  bn_stats_kernel<<<COUT, 256, 0, stream>>>(alt, cnt, MCAP, COUT, MEAN, VAR);
  bn_apply_kernel<<<((long long)MCAP * COUT + TPB - 1) / TPB, TPB, 0, stream>>>(
      alt, cg, cbe, MEAN, VAR, nullptr, alt, cnt, MCAP, COUT, 1);
  { float* t = cur; cur = alt; alt = t; }
  // two residual blocks (submanifold)
  for (int i = 0; i < 2; ++i) {
    const float* w1 = bw1 + (size_t)i * 27 * COUT * COUT;
    const float* w2 = bw2 + (size_t)i * 27 * COUT * COUT;
    conv_wmma_kernel<COUT, COUT><<<cgrid, 128, 0, stream>>>(cur, w1, subI, MCAP, cnt, MCAP, FH);
    bn_stats_kernel<<<COUT, 256, 0, stream>>>(FH, cnt, MCAP, COUT, MEAN, VAR);
    bn_apply_kernel<<<((long long)MCAP * COUT + TPB - 1) / TPB, TPB, 0, stream>>>(
        FH, bg1 + i * COUT, bbe1 + i * COUT, MEAN, VAR, nullptr, FH, cnt, MCAP, COUT, 1);
    conv_wmma_kernel<COUT, COUT><<<cgrid, 128, 0, stream>>>(FH, w2, subI, MCAP, cnt, MCAP, alt);
    bn_stats_kernel<<<COUT, 256, 0, stream>>>(alt, cnt, MCAP, COUT, MEAN, VAR);
    bn_apply_kernel<<<((long long)MCAP * COUT + TPB - 1) / TPB, TPB, 0, stream>>>(
        alt, bg2 + i * COUT, bbe2 + i * COUT, MEAN, VAR, cur, cur, cnt, MCAP, COUT, 1);
  }
  *curp = cur;
  *altp = alt;
}

// ---------------- host driver ----------------
extern "C" void kernel_launch(void* const* d_in, const int* in_sizes, int n_in,
                              void* d_out, int out_size, void* d_ws, size_t ws_size,
                              hipStream_t stream) {
  const float* vf = (const float*)d_in[0];
  const int* coors = (const int*)d_in[1];
  const float* ele_emb = (const float*)d_in[3];
  const float* azi_emb = (const float*)d_in[4];
  const float* proj_w = (const float*)d_in[5];
  const float* proj_b = (const float*)d_in[6];
  const float* wq = (const float*)d_in[7];
  const float* bq = (const float*)d_in[8];
  const float* wk = (const float*)d_in[9];
  const float* bk = (const float*)d_in[10];
  const float* wv = (const float*)d_in[11];
  const float* bv = (const float*)d_in[12];
  const float* in_w = (const float*)d_in[13];
  const float* in_b = (const float*)d_in[14];
  const float* in_g = (const float*)d_in[15];
  const float* in_be = (const float*)d_in[16];
  const float* cw[3]; const float* cg[3]; const float* cbe[3];
  const float* bw1[3]; const float* bg1[3]; const float* bbe1[3];
  const float* bw2[3]; const float* bg2[3]; const float* bbe2[3];
  for (int s = 0; s < 3; ++s) {
    int b = 17 + 9 * s;
    cw[s] = (const float*)d_in[b + 0];
    cg[s] = (const float*)d_in[b + 1];
    cbe[s] = (const float*)d_in[b + 2];
    bw1[s] = (const float*)d_in[b + 3];
    bg1[s] = (const float*)d_in[b + 4];
    bbe1[s] = (const float*)d_in[b + 5];
    bw2[s] = (const float*)d_in[b + 6];
    bg2[s] = (const float*)d_in[b + 7];
    bbe2[s] = (const float*)d_in[b + 8];
  }
  const float* out_w = (const float*)d_in[44];
  const float* out_b = (const float*)d_in[45];
  const float* out_g = (const float*)d_in[46];
  const float* out_be = (const float*)d_in[47];
  float* outp = (float*)d_out;

  char* ws = (char*)d_ws;
  size_t off = 0;
  auto alloc = [&](size_t bytes) -> void* {
    void* p = ws + off;
    off = (off + bytes + 255) & ~(size_t)255;
    return p;
  };
  float* A_ATTN = (float*)alloc((size_t)NPTS * 8 * 4);
  unsigned long long* KEYS = (unsigned long long*)alloc((size_t)4 * TSIZE * 8);
  int* VALS = (int*)alloc((size_t)4 * TSIZE * 4);
  int* CNT = (int*)alloc(64);
  int* UC0 = (int*)alloc((size_t)M0CAP * 3 * 4);
  int* UC1 = (int*)alloc((size_t)MCAP * 3 * 4);
  int* UC2 = (int*)alloc((size_t)MCAP * 3 * 4);
  int* UC3 = (int*)alloc((size_t)MCAP * 3 * 4);
  int* ROWS0 = (int*)alloc((size_t)M0CAP * 4);
  int* STR1 = (int*)alloc((size_t)27 * MCAP * 4);
  int* SUB1 = (int*)alloc((size_t)27 * MCAP * 4);
  int* STR2 = (int*)alloc((size_t)27 * MCAP * 4);
  int* SUB2 = (int*)alloc((size_t)27 * MCAP * 4);
  int* STR3 = (int*)alloc((size_t)27 * MCAP * 4);
  int* SUB3 = (int*)alloc((size_t)27 * MCAP * 4);
  float* FA = (float*)alloc((size_t)MCAP * 256 * 4);
  float* FB = (float*)alloc((size_t)MCAP * 256 * 4);
  float* FH = (float*)alloc((size_t)MCAP * 128 * 4);
  float* MEAN = (float*)alloc(256 * 4);
  float* VAR = (float*)alloc(256 * 4);
  (void)ws_size; (void)in_sizes; (void)n_in; (void)out_size;

  unsigned long long* K0 = KEYS;
  unsigned long long* K1 = KEYS + (size_t)TSIZE;
  unsigned long long* K2 = KEYS + (size_t)2 * TSIZE;
  unsigned long long* K3 = KEYS + (size_t)3 * TSIZE;
  int* V0 = VALS;
  int* V1 = VALS + (size_t)TSIZE;
  int* V2 = VALS + (size_t)2 * TSIZE;
  int* V3 = VALS + (size_t)3 * TSIZE;
  int* C0 = CNT + 0; int* C1 = CNT + 1; int* C2 = CNT + 2; int* C3 = CNT + 3;

  const int TPB = 256;
  init_tables_kernel<<<(4 * TSIZE + TPB - 1) / TPB, TPB, 0, stream>>>(KEYS, VALS, 4 * TSIZE, CNT, 4);
  attn_kernel<<<NRAYS, 256, 0, stream>>>(vf, coors, ele_emb, azi_emb, proj_w, proj_b,
                                         wq, bq, wk, bk, wv, bv, A_ATTN);
  voxelize_kernel<<<(NPTS + TPB - 1) / TPB, TPB, 0, stream>>>(coors, K0, V0);
  compact_table_kernel<<<(TSIZE + TPB - 1) / TPB, TPB, 0, stream>>>(K0, V0, 1024, 1024, UC0, ROWS0, C0, M0CAP);
  strided_insert_kernel<<<(M0CAP + TPB - 1) / TPB, TPB, 0, stream>>>(UC0, C0, M0CAP, K1, V1, 56, 512, 512);
  compact_table_kernel<<<(TSIZE + TPB - 1) / TPB, TPB, 0, stream>>>(K1, V1, 512, 512, UC1, nullptr, C1, MCAP);
  build_idx27_kernel<<<(MCAP + TPB - 1) / TPB, TPB, 0, stream>>>(UC1, C1, MCAP, K0, V0, 112, 1024, 1024, 2, STR1);
  build_idx27_kernel<<<(MCAP + TPB - 1) / TPB, TPB, 0, stream>>>(UC1, C1, MCAP, K1, V1, 56, 512, 512, 1, SUB1);
  strided_insert_kernel<<<(MCAP + TPB - 1) / TPB, TPB, 0, stream>>>(UC1, C1, MCAP, K2, V2, 28, 256, 256);
  compact_table_kernel<<<(TSIZE + TPB - 1) / TPB, TPB, 0, stream>>>(K2, V2, 256, 256, UC2, nullptr, C2, MCAP);
  build_idx27_kernel<<<(MCAP + TPB - 1) / TPB, TPB, 0, stream>>>(UC2, C2, MCAP, K1, V1, 56, 512, 512, 2, STR2);
  build_idx27_kernel<<<(MCAP + TPB - 1) / TPB, TPB, 0, stream>>>(UC2, C2, MCAP, K2, V2, 28, 256, 256, 1, SUB2);
  strided_insert_kernel<<<(MCAP + TPB - 1) / TPB, TPB, 0, stream>>>(UC2, C2, MCAP, K3, V3, 14, 128, 128);
  compact_table_kernel<<<(TSIZE + TPB - 1) / TPB, TPB, 0, stream>>>(K3, V3, 128, 128, UC3, nullptr, C3, MCAP);
  build_idx27_kernel<<<(MCAP + TPB - 1) / TPB, TPB, 0, stream>>>(UC3, C3, MCAP, K2, V2, 28, 256, 256, 2, STR3);
  build_idx27_kernel<<<(MCAP + TPB - 1) / TPB, TPB, 0, stream>>>(UC3, C3, MCAP, K3, V3, 14, 128, 128, 1, SUB3);
  in_layer_kernel<<<(M0CAP + TPB - 1) / TPB, TPB, 0, stream>>>(UC0, ROWS0, A_ATTN, in_w, in_b, in_g, in_be, C0, M0CAP, FA);

  float* cur = FA;
  float* alt = FB;
  run_stage<16, 32>(cw[0], cg[0], cbe[0], bw1[0], bg1[0], bbe1[0], bw2[0], bg2[0], bbe2[0],
                    &cur, &alt, FH, STR1, SUB1, C1, MEAN, VAR, stream);
  run_stage<32, 64>(cw[1], cg[1], cbe[1], bw1[1], bg1[1], bbe1[1], bw2[1], bg2[1], bbe2[1],
                    &cur, &alt, FH, STR2, SUB2, C2, MEAN, VAR, stream);
  run_stage<64, 128>(cw[2], cg[2], cbe[2], bw1[2], bg1[2], bbe1[2], bw2[2], bg2[2], bbe2[2],
                     &cur, &alt, FH, STR3, SUB3, C3, MEAN, VAR, stream);

  // output conv 128 -> 256 on level3 (subm), then GN(16)+relu
  {
    dim3 cgrid(MCAP / 16, 256 / 64);
    conv_wmma_kernel<128, 256><<<cgrid, 128, 0, stream>>>(cur, out_w, SUB3, MCAP, C3, MCAP, alt);
    gn_out_kernel<<<MCAP, 256, 0, stream>>>(alt, out_b, out_g, out_be, C3, MCAP);
  }
  zero_f32_kernel<<<((long long)OUTN + TPB - 1) / TPB, TPB, 0, stream>>>(outp, (long long)OUTN);
  scatter_kernel<<<MCAP, 256, 0, stream>>>(alt, UC3, C3, MCAP, outp);
}